// Model_41059887350377
// MI455X (gfx1250) — compile-verified
//
#include <hip/hip_runtime.h>

#define F_IN  512
#define F_HID 16
#define F_OUT 40

typedef __attribute__((ext_vector_type(2))) float v2f;
typedef __attribute__((ext_vector_type(8))) float v8f;

// ---------------- utility kernels ----------------

__global__ void k_zero(float* __restrict__ p, int n) {
  int i = blockIdx.x * blockDim.x + threadIdx.x;
  if (i < n) p[i] = 0.0f;
}

__global__ void k_deg(const int* __restrict__ dst, float* __restrict__ deg, int E) {
  int e = blockIdx.x * blockDim.x + threadIdx.x;
  if (e < E) atomicAdd(&deg[dst[e]], 1.0f);
}

__global__ void k_dinv(float* __restrict__ deg, int n) {
  int i = blockIdx.x * blockDim.x + threadIdx.x;
  if (i < n) deg[i] = rsqrtf(deg[i] + 1.0f);
}

// ---------------- GEMM1: h1[N,16] = x[N,512] @ W1[512,16] (WMMA f32 16x16x4) ----

__global__ __launch_bounds__(256) void k_gemm1(const float* __restrict__ x,
                                               const float* __restrict__ W1,
                                               float* __restrict__ h1,
                                               int ntiles) {
  int wave = (blockIdx.x * blockDim.x + threadIdx.x) >> 5;
  if (wave >= ntiles) return;        // wave-uniform exit; EXEC stays all-1s
  int lane = threadIdx.x & 31;
  int hi   = lane >> 4;              // 0: lanes 0-15, 1: lanes 16-31
  int l    = lane & 15;

  const float* __restrict__ xrow = x + (size_t)(wave * 16 + l) * F_IN;

  v8f acc = {};
#pragma unroll 4
  for (int k = 0; k < F_IN; k += 4) {
    int kk = k + 2 * hi;
    // A 16x4 f32: VGPR0 = A[m][kk], VGPR1 = A[m][kk+1]
    v2f a = *(const v2f*)(xrow + kk);               // 8B aligned (kk even, row 2KB aligned)
    // B 4x16 f32: VGPR0 = B[kk][n], VGPR1 = B[kk+1][n], n = lane&15
    v2f b;
    b.x = W1[(size_t)kk       * F_HID + l];
    b.y = W1[(size_t)(kk + 1) * F_HID + l];
    acc = __builtin_amdgcn_wmma_f32_16x16x4_f32(false, a, false, b,
                                                (short)0, acc, false, false);
  }

  // C/D: VGPR r -> M = r (lanes 0-15) or 8+r (lanes 16-31), N = lane&15
  size_t base = (size_t)wave * 16;
#pragma unroll
  for (int r = 0; r < 8; ++r) {
    h1[(base + r + 8 * hi) * F_HID + l] = acc[r];
  }
}

// ---------------- layer-1 edge aggregation ----------------

__global__ void k_edge1(const int* __restrict__ src, const int* __restrict__ dst,
                        const float* __restrict__ dinv, const float* __restrict__ h1,
                        float* __restrict__ agg, int E) {
  int e = blockIdx.x * blockDim.x + threadIdx.x;
  if (e >= E) return;
  int s = src[e], d = dst[e];
  float c = dinv[s] * dinv[d];
  const float4* __restrict__ hs = (const float4*)(h1 + (size_t)s * F_HID);
  float* __restrict__ o = agg + (size_t)d * F_HID;
#pragma unroll
  for (int q = 0; q < F_HID / 4; ++q) {
    float4 v = hs[q];
    atomicAdd(&o[q * 4 + 0], v.x * c);
    atomicAdd(&o[q * 4 + 1], v.y * c);
    atomicAdd(&o[q * 4 + 2], v.z * c);
    atomicAdd(&o[q * 4 + 3], v.w * c);
  }
}

// finalize: agg += h1*dinv^2 + b1, then ReLU (in place -> becomes h_relu)
__global__ void k_fin1(const float* __restrict__ h1, const float* __restrict__ dinv,
                       const float* __restrict__ b1, float* __restrict__ agg, int total) {
  int idx = blockIdx.x * blockDim.x + threadIdx.x;
  if (idx >= total) return;
  int i = idx >> 4;
  int f = idx & (F_HID - 1);
  float di = dinv[i];
  float v = agg[idx] + h1[idx] * di * di + b1[f];
  agg[idx] = v > 0.0f ? v : 0.0f;
}

// ---------------- GEMM2: h2[N,40] = h[N,16] @ W2[16,40] (3 padded N-tiles) ----

__global__ __launch_bounds__(256) void k_gemm2(const float* __restrict__ h,
                                               const float* __restrict__ W2,
                                               float* __restrict__ h2,
                                               int ntiles) {
  int wave = (blockIdx.x * blockDim.x + threadIdx.x) >> 5;
  if (wave >= ntiles) return;
  int lane = threadIdx.x & 31;
  int hi   = lane >> 4;
  int l    = lane & 15;

  const float* __restrict__ hrow = h + (size_t)(wave * 16 + l) * F_HID;

  v8f acc0 = {}, acc1 = {}, acc2 = {};
#pragma unroll
  for (int k = 0; k < F_HID; k += 4) {
    int kk = k + 2 * hi;
    v2f a = *(const v2f*)(hrow + kk);
    int n0 = l, n1 = 16 + l, n2 = 32 + l;
    v2f b0, b1, b2;
    b0.x = W2[(size_t)kk       * F_OUT + n0];
    b0.y = W2[(size_t)(kk + 1) * F_OUT + n0];
    b1.x = W2[(size_t)kk       * F_OUT + n1];
    b1.y = W2[(size_t)(kk + 1) * F_OUT + n1];
    b2.x = (n2 < F_OUT) ? W2[(size_t)kk       * F_OUT + n2] : 0.0f;
    b2.y = (n2 < F_OUT) ? W2[(size_t)(kk + 1) * F_OUT + n2] : 0.0f;
    acc0 = __builtin_amdgcn_wmma_f32_16x16x4_f32(false, a, false, b0, (short)0, acc0, false, false);
    acc1 = __builtin_amdgcn_wmma_f32_16x16x4_f32(false, a, false, b1, (short)0, acc1, false, false);
    acc2 = __builtin_amdgcn_wmma_f32_16x16x4_f32(false, a, false, b2, (short)0, acc2, false, false);
  }

  size_t base = (size_t)wave * 16;
#pragma unroll
  for (int r = 0; r < 8; ++r) {
    size_t m = base + r + 8 * hi;
    h2[m * F_OUT + l]      = acc0[r];
    h2[m * F_OUT + 16 + l] = acc1[r];
    if (32 + l < F_OUT) h2[m * F_OUT + 32 + l] = acc2[r];
  }
}

// ---------------- layer-2 self term + bias into d_out ----------------

__global__ void k_init_out(const float* __restrict__ h2, const float* __restrict__ dinv,
                           const float* __restrict__ b2, float* __restrict__ out, int total) {
  int idx = blockIdx.x * blockDim.x + threadIdx.x;
  if (idx >= total) return;
  int i = idx / F_OUT;
  int f = idx - i * F_OUT;
  float di = dinv[i];
  out[idx] = h2[idx] * di * di + b2[f];
}

// ---------------- layer-2 edge aggregation ----------------

__global__ void k_edge2(const int* __restrict__ src, const int* __restrict__ dst,
                        const float* __restrict__ dinv, const float* __restrict__ h2,
                        float* __restrict__ out, int E) {
  int e = blockIdx.x * blockDim.x + threadIdx.x;
  if (e >= E) return;
  int s = src[e], d = dst[e];
  float c = dinv[s] * dinv[d];
  const float4* __restrict__ hs = (const float4*)(h2 + (size_t)s * F_OUT);
  float* __restrict__ o = out + (size_t)d * F_OUT;
#pragma unroll
  for (int q = 0; q < F_OUT / 4; ++q) {
    float4 v = hs[q];
    atomicAdd(&o[q * 4 + 0], v.x * c);
    atomicAdd(&o[q * 4 + 1], v.y * c);
    atomicAdd(&o[q * 4 + 2], v.z * c);
    atomicAdd(&o[q * 4 + 3], v.w * c);
  }
}

// ---------------- launch ----------------

extern "C" void kernel_launch(void* const* d_in, const int* in_sizes, int n_in,
                              void* d_out, int out_size, void* d_ws, size_t ws_size,
                              hipStream_t stream) {
  const float* x  = (const float*)d_in[0];
  const int*   ei = (const int*)d_in[1];   // [2, E] int32, row-major: src then dst
  const float* W1 = (const float*)d_in[2];
  const float* b1 = (const float*)d_in[3];
  const float* W2 = (const float*)d_in[4];
  const float* b2 = (const float*)d_in[5];
  float* out = (float*)d_out;

  const int N = in_sizes[0] / F_IN;
  const int E = in_sizes[1] / 2;
  const int* src = ei;
  const int* dst = ei + E;

  float* ws   = (float*)d_ws;
  float* dinv = ws;                            // N       (first used as deg)
  float* h1   = dinv + N;                      // N*16
  float* agg1 = h1 + (size_t)N * F_HID;        // N*16    (becomes h_relu in place)
  float* h2   = agg1 + (size_t)N * F_HID;      // N*40

  const int B = 256;
  const int ntiles = (N + 15) / 16;            // 6250 (N is a multiple of 16)

  // degree + dinv
  k_zero<<<(N + B - 1) / B, B, 0, stream>>>(dinv, N);
  k_deg<<<(E + B - 1) / B, B, 0, stream>>>(dst, dinv, E);
  k_dinv<<<(N + B - 1) / B, B, 0, stream>>>(dinv, N);

  // layer 1
  k_gemm1<<<(ntiles * 32 + B - 1) / B, B, 0, stream>>>(x, W1, h1, ntiles);
  k_zero<<<(N * F_HID + B - 1) / B, B, 0, stream>>>(agg1, N * F_HID);
  k_edge1<<<(E + B - 1) / B, B, 0, stream>>>(src, dst, dinv, h1, agg1, E);
  k_fin1<<<(N * F_HID + B - 1) / B, B, 0, stream>>>(h1, dinv, b1, agg1, N * F_HID);

  // layer 2
  k_gemm2<<<(ntiles * 32 + B - 1) / B, B, 0, stream>>>(agg1, W2, h2, ntiles);
  k_init_out<<<(N * F_OUT + B - 1) / B, B, 0, stream>>>(h2, dinv, b2, out, N * F_OUT);
  k_edge2<<<(E + B - 1) / B, B, 0, stream>>>(src, dst, dinv, h2, out, E);
}